// RelationBasedTransform_89850715833077
// MI455X (gfx1250) — compile-verified
//
#include <hip/hip_runtime.h>

// Problem constants (from the reference)
#define B_SAMPLES 4096
#define D_INF     300
#define D_OUTF    300
#define NUM_MAT   274
#define TILE      16
#define NT_OUT    ((D_OUTF + TILE - 1) / TILE)          // 19 out-row tiles
#define MAX_TILES (NUM_MAT + (B_SAMPLES / TILE))        // <= 530 sample tiles

typedef __attribute__((ext_vector_type(2))) float v2f;
typedef __attribute__((ext_vector_type(4))) float v4f;
typedef __attribute__((ext_vector_type(8))) float v8f;

// ---------------------------------------------------------------------------
// Pass 1: zero the workspace counters
// ---------------------------------------------------------------------------
__global__ void rbt_init(int* __restrict__ counts, int* __restrict__ cursor,
                         int* __restrict__ ntiles) {
    int i = blockIdx.x * blockDim.x + threadIdx.x;
    if (i < NUM_MAT) { counts[i] = 0; cursor[i] = 0; }
    if (i == 0) { *ntiles = 0; }
}

// ---------------------------------------------------------------------------
// Pass 2: histogram of relation ids
// ---------------------------------------------------------------------------
__global__ void rbt_hist(const int* __restrict__ rel, int* __restrict__ counts) {
    int b = blockIdx.x * blockDim.x + threadIdx.x;
    if (b < B_SAMPLES) atomicAdd(&counts[rel[b]], 1);
}

// ---------------------------------------------------------------------------
// Pass 3: serial prefix scan over 274 relations + build tile descriptors.
// 274 iterations on one lane: negligible vs the memory-bound main pass.
// ---------------------------------------------------------------------------
__global__ void rbt_scan(const int* __restrict__ counts, int* __restrict__ offsets,
                         int* __restrict__ tiles, int* __restrict__ ntiles) {
    if (threadIdx.x != 0 || blockIdx.x != 0) return;
    int acc = 0, t = 0;
    for (int r = 0; r < NUM_MAT; ++r) {
        offsets[r] = acc;
        int c = counts[r];
        for (int s = 0; s < c; s += TILE) {
            tiles[3 * t + 0] = r;
            tiles[3 * t + 1] = acc + s;
            tiles[3 * t + 2] = (c - s < TILE) ? (c - s) : TILE;
            ++t;
        }
        acc += c;
    }
    offsets[NUM_MAT] = acc;
    *ntiles = t;
}

// ---------------------------------------------------------------------------
// Pass 4: scatter sample ids into relation-sorted order[]
// ---------------------------------------------------------------------------
__global__ void rbt_scatter(const int* __restrict__ rel, const int* __restrict__ offsets,
                            int* __restrict__ cursor, int* __restrict__ order) {
    int b = blockIdx.x * blockDim.x + threadIdx.x;
    if (b < B_SAMPLES) {
        int r = rel[b];
        int pos = offsets[r] + atomicAdd(&cursor[r], 1);
        order[pos] = b;
    }
}

// ---------------------------------------------------------------------------
// Pass 5: grouped GEMM. One wave32 per (out-row-tile, sample-tile) pair.
// D(16x16,f32) = A(16x4,f32: rows of W[r]) x B(4x16,f32: gathered x columns)
// accumulated over K = 300 in steps of 4 with V_WMMA_F32_16X16X4_F32.
//
// FP32 16x16x4 operand layout (ISA 7.12.2):
//   A: lanes 0-15 -> M=lane,  v0=K0 v1=K1 ; lanes 16-31 -> M=lane-16, v0=K2 v1=K3
//   B: lanes 0-15 -> N=lane,  v0=K0 v1=K1 ; lanes 16-31 -> N=lane-16, v0=K2 v1=K3
//   D: lanes 0-15 -> N=lane, VGPR v = M=v ; lanes 16-31 -> N=lane-16, VGPR v = M=8+v
// All control flow before the stores is wave-uniform or clamp-based, so EXEC
// stays all-1s through every WMMA (ISA requirement).
// ---------------------------------------------------------------------------
__global__ void rbt_gemm(const float* __restrict__ x, const float* __restrict__ W,
                         float* __restrict__ out, const int* __restrict__ order,
                         const int* __restrict__ tiles, const int* __restrict__ ntiles) {
    const int tid = blockIdx.y;
    if (tid >= *ntiles) return;                      // wave-uniform early exit

    const int r     = tiles[3 * tid + 0];
    const int start = tiles[3 * tid + 1];
    const int cnt   = tiles[3 * tid + 2];

    const int lane = threadIdx.x;                    // 0..31 (wave32)
    const int half = lane >> 4;                      // 0: K0/K1, 1: K2/K3
    const int l16  = lane & 15;

    const int rowbase = blockIdx.x * TILE;           // first out-row of this tile
    const int arow = (rowbase + l16 < D_OUTF) ? (rowbase + l16) : (D_OUTF - 1); // clamp, store-guarded
    const int col  = (l16 < cnt) ? l16 : (cnt - 1);                              // clamp, store-guarded
    const int s    = order[start + col];

    // 8-byte aligned float2 streams: row strides (300 floats) and K offsets are even.
    const float* wp = W + (size_t)r * (D_OUTF * D_INF) + (size_t)arow * D_INF + 2 * half;
    const float* xp = x + (size_t)s * D_INF + 2 * half;

    v8f c = {};
#pragma unroll 5
    for (int k = 0; k < D_INF; k += 4) {
        v2f a = *(const v2f*)(wp + k);               // A: K=k+2*half .. +1
        v2f b = *(const v2f*)(xp + k);               // B: K=k+2*half .. +1
        c = __builtin_amdgcn_wmma_f32_16x16x4_f32(
                /*neg_a=*/false, a, /*neg_b=*/false, b,
                /*c_mod=*/(short)0, c, /*reuse_a=*/false, /*reuse_b=*/false);
    }

    // D: each lane owns 8 consecutive out-dim elements of one sample's row.
    if (l16 < cnt) {
        const int m0 = rowbase + 8 * half;
        float* op = out + (size_t)s * D_OUTF + m0;   // 16B aligned (1200*s + 64*bx + 32*half)
        if (m0 + 8 <= D_OUTF) {
            v4f lo = {c[0], c[1], c[2], c[3]};
            v4f hi = {c[4], c[5], c[6], c[7]};
            *(v4f*)(op + 0) = lo;                    // global_store_b128
            *(v4f*)(op + 4) = hi;                    // global_store_b128
        } else {
#pragma unroll
            for (int v = 0; v < 8; ++v)
                if (m0 + v < D_OUTF) op[v] = c[v];   // tail tile (rows 296..299)
        }
    }
}

// ---------------------------------------------------------------------------
extern "C" void kernel_launch(void* const* d_in, const int* in_sizes, int n_in,
                              void* d_out, int out_size, void* d_ws, size_t ws_size,
                              hipStream_t stream) {
    const float* x   = (const float*)d_in[0];        // [4096,300] f32
    const int*   rel = (const int*)  d_in[1];        // [4096] i32
    const float* W   = (const float*)d_in[2];        // [274,300,300] f32
    float*       out = (float*)d_out;                // [4096,300] f32

    // Workspace layout (ints): ~26 KB total
    int* ws      = (int*)d_ws;
    int* counts  = ws;                               // [274]
    int* offsets = counts + NUM_MAT;                 // [275]
    int* cursor  = offsets + NUM_MAT + 1;            // [274]
    int* order   = cursor + NUM_MAT;                 // [4096]
    int* ntiles  = order + B_SAMPLES;                // [1]
    int* tiles   = ntiles + 1;                       // [MAX_TILES*3]

    rbt_init   <<<1, 320, 0, stream>>>(counts, cursor, ntiles);
    rbt_hist   <<<B_SAMPLES / 256, 256, 0, stream>>>(rel, counts);
    rbt_scan   <<<1, 32, 0, stream>>>(counts, offsets, tiles, ntiles);
    rbt_scatter<<<B_SAMPLES / 256, 256, 0, stream>>>(rel, offsets, cursor, order);

    dim3 grid(NT_OUT, MAX_TILES);                    // (19, 530); extra tiles exit early
    rbt_gemm   <<<grid, 32, 0, stream>>>(x, W, out, order, tiles, ntiles);
}